// GMMGrid_90795608638128
// MI455X (gfx1250) — compile-verified
//
#include <hip/hip_runtime.h>
#include <stdint.h>

#define SQRT2f 1.41421356237309515f
#define NCNf  -0.91893853320467274f
#define EPSf   1e-30f

// ---- gfx1250 async global->LDS staging (ASYNCcnt path) ------------------
#if defined(__gfx1250__) && __has_builtin(__builtin_amdgcn_global_load_async_to_lds_b32)
#define USE_ASYNC 1
#else
#define USE_ASYNC 0
#endif

__device__ __forceinline__ void async_copy_f32(const float* gsrc, float* lds_dst) {
#if USE_ASYNC
    __builtin_amdgcn_global_load_async_to_lds_b32(
        (__attribute__((address_space(1))) int*)(uintptr_t)gsrc,
        (__attribute__((address_space(3))) int*)(uint32_t)(uintptr_t)lds_dst,
        0, 0);
#else
    *lds_dst = *gsrc;
#endif
}

__device__ __forceinline__ void wait_async_all() {
#if USE_ASYNC
#if __has_builtin(__builtin_amdgcn_s_wait_asynccnt)
    __builtin_amdgcn_s_wait_asynccnt(0);
#else
    asm volatile("s_wait_asynccnt 0" ::: "memory");
#endif
#endif
}

// Output layout (flat float offsets, C-order concat of the return tuple)
//  xn   : (4,32,32,1,4,17)      ->      278528 @ 0
//  xe   : (4,32,32,2,4,289,2)   ->    18939904 @ 278528
//  lpn  : (4,32,32,1,4,17)      ->      278528 @ 19218432
//  lpe  : (4,32,32,2,4,289)     ->     9469952 @ 19496960
//  alpha: (4,1,1,1,4,1)         ->          16 @ 28966912
//  z1   : (4,32,32,2,4,289)     ->     9469952 @ 28966928
//  z2   : (4,32,32,2,4,289)     ->     9469952 @ 38436880
//  o2   : (4,32,32,2,4,1)       ->       32768 @ 47906832
#define OFF_XN    0LL
#define OFF_XE    278528LL
#define OFF_LPN   19218432LL
#define OFF_LPE   19496960LL
#define OFF_ALPHA 28966912LL
#define OFF_Z1    28966928LL
#define OFF_Z2    38436880LL
#define OFF_O2    47906832LL

__global__ __launch_bounds__(256)
void gmmgrid_kernel(const float* __restrict__ mu, const float* __restrict__ sigma,
                    const float* __restrict__ rou, const float* __restrict__ w,
                    const float* __restrict__ xq, const float* __restrict__ xi,
                    const float* __restrict__ xj, float* __restrict__ out)
{
    constexpr int K = 17, K2 = 289;

    __shared__ float s_xi[K2], s_xj[K2], s_xq[K];
    __shared__ float s_mu[4], s_sig[4], s_w[4], s_rou[8];
    __shared__ float s_mu2[8], s_sig2[8];
    // derived per-site invariants
    __shared__ float s_alpha[4], s_isig[4], s_cn[4];
    __shared__ float s_a[8], s_b[8], s_p[8], s_i2q[8], s_ce[8], s_isig2[8];

    const int tid  = threadIdx.x;
    const int site = blockIdx.x;            // ((c*32)+m)*32+n
    const int cc   = site >> 10;
    const int rem  = site & 1023;
    const int mm   = rem >> 5;
    const int nn   = rem & 31;
    const int siteM = (cc << 10) | (((mm + 1) & 31) << 5) | nn;   // roll m by -1
    const int siteN = (cc << 10) | (mm << 5) | ((nn + 1) & 31);   // roll n by -1

    // ---- stage Hermite tables via async global->LDS copies ----
    if (tid < K)        async_copy_f32(xq + tid, s_xq + tid);
    {   // xi/xj: 289 each, two rounds of 256 threads
        async_copy_f32(xi + tid, s_xi + tid);
        async_copy_f32(xj + tid, s_xj + tid);
        if (tid < K2 - 256) {
            async_copy_f32(xi + 256 + tid, s_xi + 256 + tid);
            async_copy_f32(xj + 256 + tid, s_xj + 256 + tid);
        }
    }
    // ---- per-site parameters (36 scalars) via normal loads ----
    if (tid < 4) {
        s_mu[tid]  = mu[site * 4 + tid];
        s_sig[tid] = sigma[site * 4 + tid];
        s_w[tid]   = w[cc * 4 + tid];
    } else if (tid < 12) {
        int t = tid - 4;
        s_rou[t] = rou[site * 8 + t];
    } else if (tid < 16) {
        int l = tid - 12;
        s_mu2[l]  = mu[siteM * 4 + l];
        s_sig2[l] = sigma[siteM * 4 + l];
    } else if (tid < 20) {
        int l = tid - 16;
        s_mu2[4 + l]  = mu[siteN * 4 + l];
        s_sig2[4 + l] = sigma[siteN * 4 + l];
    }

    wait_async_all();
    __syncthreads();

    // ---- one thread precomputes invariants (cheap vs 2312-item loop) ----
    if (tid == 0) {
        float wm = fmaxf(fmaxf(s_w[0], s_w[1]), fmaxf(s_w[2], s_w[3]));
        float e[4], sum = 0.f;
        #pragma unroll
        for (int l = 0; l < 4; ++l) { e[l] = __expf(s_w[l] - wm); sum += e[l]; }
        float inv = 1.f / sum;
        #pragma unroll
        for (int l = 0; l < 4; ++l) {
            s_alpha[l] = e[l] * inv;
            s_isig[l]  = 1.f / s_sig[l];
            s_cn[l]    = s_alpha[l] * s_isig[l];
        }
        #pragma unroll
        for (int jl = 0; jl < 8; ++jl) {
            float p  = s_rou[jl];
            float qv = 1.f - p * p;
            float is2 = 1.f / s_sig2[jl];
            s_p[jl]     = p;
            s_i2q[jl]   = 1.f / (2.f * qv);
            s_isig2[jl] = is2;
            s_ce[jl]    = s_alpha[jl & 3] * s_isig[jl & 3] * is2 * rsqrtf(qv);
            float sp = sqrtf(1.f + p) * 0.5f;
            float tp = sqrtf(1.f - p) * 0.5f;
            s_a[jl] = sp + tp;
            s_b[jl] = sp - tp;
        }
    }
    __syncthreads();

    // ---- tiny outputs ----
    if (rem == 0 && tid < 4) out[OFF_ALPHA + cc * 4 + tid] = s_alpha[tid];
    if (tid < 8)             out[OFF_O2 + (long long)site * 8 + tid] = s_sig2[tid];

    // ---- xn / lpn : 68 items ----
    if (tid < 68) {
        int l = tid / 17, q = tid % 17;
        float xnv = fmaf(s_xq[q] * SQRT2f, s_sig[l], s_mu[l]);
        out[OFF_XN + (long long)site * 68 + tid] = xnv;
        float acc = 0.f;
        #pragma unroll
        for (int lp = 0; lp < 4; ++lp) {
            float d = (xnv - s_mu[lp]) * s_isig[lp];
            acc = fmaf(__expf(-0.5f * d * d), s_cn[lp], acc);
        }
        out[OFF_LPN + (long long)site * 68 + tid] = __logf(acc + EPSf) + NCNf;
    }

    // ---- z1/z2/xe/lpe : 2312 items, [jl][q] with jl = j*4+l ----
    const long long base1  = (long long)site * 2312;
    const long long basexe = (long long)site * 4624;
    for (int i = tid; i < 2312; i += 256) {
        int q  = i % K2;
        int jl = i / K2;
        int l  = jl & 3;
        int j4 = jl & 4;   // j*4
        float XI = s_xi[q], XJ = s_xj[q];
        float av = s_a[jl], bv = s_b[jl];
        float z1 = fmaf(av, XI, bv * XJ);
        float z2 = fmaf(bv, XI, av * XJ);
        float x1 = fmaf(z1 * SQRT2f, s_sig[l],  s_mu[l]);
        float x2 = fmaf(z2 * SQRT2f, s_sig2[jl], s_mu2[jl]);
        out[OFF_Z1 + base1 + i] = z1;
        out[OFF_Z2 + base1 + i] = z2;
        *(float2*)(out + OFF_XE + basexe + 2 * i) = make_float2(x1, x2);

        float acc = 0.f;
        #pragma unroll
        for (int lp = 0; lp < 4; ++lp) {
            int jlp = j4 + lp;
            float d1 = (x1 - s_mu[lp])  * s_isig[lp];
            float d2 = (x2 - s_mu2[jlp]) * s_isig2[jlp];
            float zz = -(d1 * d1 - 2.f * s_p[jlp] * d1 * d2 + d2 * d2) * s_i2q[jlp];
            acc = fmaf(__expf(zz), s_ce[jlp], acc);
        }
        out[OFF_LPE + base1 + i] = __logf(acc + EPSf) + 2.f * NCNf;
    }
}

extern "C" void kernel_launch(void* const* d_in, const int* in_sizes, int n_in,
                              void* d_out, int out_size, void* d_ws, size_t ws_size,
                              hipStream_t stream) {
    const float* mu    = (const float*)d_in[0];
    const float* sigma = (const float*)d_in[1];
    const float* rou   = (const float*)d_in[2];
    const float* w     = (const float*)d_in[3];
    const float* xq    = (const float*)d_in[4];
    const float* xi    = (const float*)d_in[5];
    const float* xj    = (const float*)d_in[6];
    float* out = (float*)d_out;
    (void)in_sizes; (void)n_in; (void)out_size; (void)d_ws; (void)ws_size;

    dim3 grid(4096), block(256);
    hipLaunchKernelGGL(gmmgrid_kernel, grid, block, 0, stream,
                       mu, sigma, rou, w, xq, xi, xj, out);
}